// HebbFF_22789096472917
// MI455X (gfx1250) — compile-verified
//
#include <hip/hip_runtime.h>

#define HID 4096
#define INP 2048
#define T_LEN 1024
#define NWG 512          // workgroups owning 8 rows each: 512*8 = 4096
#define KPL 64           // columns of M per lane: 2048 / 32 lanes

typedef float v2f __attribute__((ext_vector_type(2)));
typedef float v8f __attribute__((ext_vector_type(8)));

// ---------------------------------------------------------------------------
// Kernel A: PA[t, i] = sum_j X[t, j] * wa[i, j]   (time-invariant term, WMMA)
// f32 16x16x4 WMMA, 4x4 register blocking: each wave owns a 64(t) x 64(i)
// strip -> 16 accumulator tiles; per K-chunk 8 loads feed 16 WMMAs (4x reuse,
// ~256 B of L2 traffic per WMMA instead of ~1 KB).
// Lane layout per ISA: lanes 0-15 hold K=0,1; lanes 16-31 hold K=2,3.
// D: VGPR r holds rows m=r (lanes 0-15) and m=r+8 (lanes 16-31), n = lane%16.
// ---------------------------------------------------------------------------
__global__ __launch_bounds__(256) void wa_gemm_wmma(
    const float* __restrict__ X,    // [T, INP]
    const float* __restrict__ wa,   // [HID, INP]
    float* __restrict__ PA)         // [T, HID]
{
    const int wave = threadIdx.x >> 5;
    const int lane = threadIdx.x & 31;
    const int half = lane >> 4;     // 0: K=0,1   1: K=2,3
    const int sub  = lane & 15;

    // 1024 waves: 64 i-strips x 16 t-strips, each strip 64x64
    const int w  = blockIdx.x * 8 + wave;
    const int si = w & 63;          // i-strip
    const int st = w >> 6;          // t-strip
    const int i0 = si * 64;
    const int t0 = st * 64;

    // Per-lane row base pointers, with the K sub-offset (2*half) folded in.
    const float* ax[4];
    const float* bw[4];
    #pragma unroll
    for (int j = 0; j < 4; ++j) {
        ax[j] = X  + (size_t)(t0 + j * 16 + sub) * INP + 2 * half;
        bw[j] = wa + (size_t)(i0 + j * 16 + sub) * INP + 2 * half;
    }

    v8f c[4][4];
    #pragma unroll
    for (int jt = 0; jt < 4; ++jt)
        #pragma unroll
        for (int ji = 0; ji < 4; ++ji)
            c[jt][ji] = (v8f){0.f, 0.f, 0.f, 0.f, 0.f, 0.f, 0.f, 0.f};

    #pragma unroll 2
    for (int kk = 0; kk < INP; kk += 4) {
        v2f a[4], b[4];
        #pragma unroll
        for (int j = 0; j < 4; ++j) a[j] = *(const v2f*)(ax[j] + kk);
        #pragma unroll
        for (int j = 0; j < 4; ++j) b[j] = *(const v2f*)(bw[j] + kk);
        #pragma unroll
        for (int jt = 0; jt < 4; ++jt)
            #pragma unroll
            for (int ji = 0; ji < 4; ++ji)
                c[jt][ji] = __builtin_amdgcn_wmma_f32_16x16x4_f32(
                    false, a[jt], false, b[ji], (short)0, c[jt][ji],
                    false, false);
    }

    #pragma unroll
    for (int jt = 0; jt < 4; ++jt)
        #pragma unroll
        for (int ji = 0; ji < 4; ++ji)
            #pragma unroll
            for (int r = 0; r < 8; ++r) {
                const int m = r + 8 * half;
                PA[(size_t)(t0 + jt * 16 + m) * HID + i0 + ji * 16 + sub] =
                    c[jt][ji][r];                 // coalesced over sub
            }
}

// ---------------------------------------------------------------------------
// Kernel B: the sequential recurrence. Each workgroup owns 8 rows; each wave
// owns one row; each lane keeps 64 M-elements + 64 (wm+1)-elements in VGPRs
// for all T=1024 steps. x_t is double-buffered in LDS and the x_{t+1} global
// loads are issued before step t's compute to hide L2 latency.
// ---------------------------------------------------------------------------
__global__ __launch_bounds__(256) void hebb_recurrence(
    const float* __restrict__ X,          // [T, INP]
    const float* __restrict__ wm,         // [HID, INP]
    const float* __restrict__ b1,         // [HID]
    const float* __restrict__ w2,         // [1, HID]
    const float* __restrict__ M0,         // [HID, INP]
    const float* __restrict__ PA,         // [T, HID]
    const float* __restrict__ lmbda_mult, // scalar
    const float* __restrict__ eta_mult,   // scalar
    float* __restrict__ partials)         // [NWG, T]
{
    __shared__ float xs[2][INP];          // 16 KB double buffer
    __shared__ float hpart[8];

    const int tid  = threadIdx.x;
    const int wave = tid >> 5;
    const int lane = tid & 31;
    const int row  = blockIdx.x * 8 + wave;

    const float decay = 1.0f / (1.0f + __expf(-lmbda_mult[0]));
    const float eta   = eta_mult[0];
    const float b1r   = b1[row];
    const float w2r   = w2[row];

    // Pin this row's M slice and (wm+1) slice in registers (coalesced loads).
    float m[KPL], w[KPL];
    const size_t base = (size_t)row * INP + lane;
    #pragma unroll
    for (int k = 0; k < KPL; ++k) {
        m[k] = M0[base + 32 * k];
        w[k] = wm[base + 32 * k] + 1.0f;
    }

    // Preload x_0 into registers.
    float xr[INP / 256];
    #pragma unroll
    for (int j = 0; j < INP / 256; ++j)
        xr[j] = X[j * 256 + tid];

    int buf = 0;
    for (int t = 0; t < T_LEN; ++t) {
        // publish x_t (registers -> LDS buffer `buf`)
        #pragma unroll
        for (int j = 0; j < INP / 256; ++j)
            xs[buf][j * 256 + tid] = xr[j];
        __syncthreads();                  // barrier A: xs[buf] ready

        // issue x_{t+1} loads now; s_wait lands at next iteration's publish
        if (t + 1 < T_LEN) {
            #pragma unroll
            for (int j = 0; j < INP / 256; ++j)
                xr[j] = X[(size_t)(t + 1) * INP + j * 256 + tid];
            if (t + 2 < T_LEN)
                __builtin_prefetch(&X[(size_t)(t + 2) * INP + tid * 8], 0, 0);
        }

        // pre[row] partial: sum_j (wm+1)*M*x over this lane's 64 columns
        float acc = 0.f;
        #pragma unroll
        for (int k = 0; k < KPL; ++k)
            acc = fmaf(w[k] * m[k], xs[buf][32 * k + lane], acc);
        #pragma unroll
        for (int off = 16; off >= 1; off >>= 1)
            acc += __shfl_xor(acc, off, 32);   // wave32 xor-reduce (all lanes)

        const float pre = acc + PA[(size_t)t * HID + row] + b1r;
        const float h   = 1.0f / (1.0f + __expf(-pre));

        // Hebbian write: M = decay*M + eta*h*x (row-local, register resident)
        const float eh = eta * h;
        #pragma unroll
        for (int k = 0; k < KPL; ++k)
            m[k] = fmaf(eh, xs[buf][32 * k + lane], decay * m[k]);

        if (lane == 0) hpart[wave] = w2r * h;
        __syncthreads();                  // barrier B: hpart ready, xs reads done
        if (tid == 0) {
            float s = 0.f;
            #pragma unroll
            for (int v = 0; v < 8; ++v) s += hpart[v];
            partials[(size_t)blockIdx.x * T_LEN + t] = s;
        }
        buf ^= 1;
        // xs[buf^1] (just read in step t) is next written in step t+2, with
        // barriers A(t+1) and B(t+1) in between -> no WAR hazard.
    }
}

// ---------------------------------------------------------------------------
// Kernel C: y[t] = sigmoid(b2 + sum_g partials[g][t])  (deterministic order)
// ---------------------------------------------------------------------------
__global__ __launch_bounds__(256) void finalize_y(
    const float* __restrict__ partials,   // [NWG, T]
    const float* __restrict__ b2,
    float* __restrict__ y)                // [T]
{
    const int t = blockIdx.x * blockDim.x + threadIdx.x;
    if (t < T_LEN) {
        float s = b2[0];
        #pragma unroll 8
        for (int g = 0; g < NWG; ++g)
            s += partials[(size_t)g * T_LEN + t];   // coalesced across t
        y[t] = 1.0f / (1.0f + __expf(-s));
    }
}

extern "C" void kernel_launch(void* const* d_in, const int* in_sizes, int n_in,
                              void* d_out, int out_size, void* d_ws, size_t ws_size,
                              hipStream_t stream) {
    const float* x_seq = (const float*)d_in[0];
    const float* wa    = (const float*)d_in[1];
    const float* wm    = (const float*)d_in[2];
    const float* b1    = (const float*)d_in[3];
    const float* w2    = (const float*)d_in[4];
    const float* b2    = (const float*)d_in[5];
    const float* lm    = (const float*)d_in[6];
    const float* em    = (const float*)d_in[7];
    const float* M0    = (const float*)d_in[8];
    float* y = (float*)d_out;

    float* PA       = (float*)d_ws;                 // T*HID floats (16.8 MB)
    float* partials = PA + (size_t)T_LEN * HID;     // NWG*T floats (2 MB)

    // A: hoisted time-invariant GEMM PA = X @ wa^T (WMMA f32, 4x4 blocking)
    // 1024 waves = 64 i-strips x 16 t-strips -> 128 blocks of 8 waves
    wa_gemm_wmma<<<128, 256, 0, stream>>>(x_seq, wa, PA);
    // B: sequential fast-weight recurrence, M + (wm+1) resident in VGPRs
    hebb_recurrence<<<NWG, 256, 0, stream>>>(x_seq, wm, b1, w2, M0, PA, lm, em, partials);
    // C: reduce per-WG y contributions
    finalize_y<<<(T_LEN + 255) / 256, 256, 0, stream>>>(partials, b2, y);
}